// HashEmbedder_43387759624288
// MI455X (gfx1250) — compile-verified
//
#include <hip/hip_runtime.h>
#include <math.h>

// Instant-NGP hash-grid forward for MI455X (gfx1250, wave32).
// Gather-bound kernel: stage hot coarse-level tables (levels 0..5, ~107 KB)
// into LDS via CDNA5 async global->LDS DMA (ASYNCcnt), gather them with
// ds_load_b64; fine/hashed levels gather from L2-resident tables with
// global_load_b64; 8x global_store_b128 per point.

#define N_LEVELS 16
#define N_STAGED 6            // levels 0..5 fit in SMEM_BYTES; all direct-indexed
#define LOG2T 19
#define TMASK ((1u << LOG2T) - 1u)
#define BLOCK 256
#define PPT 4                 // points per thread
#define SMEM_BYTES 110592     // ~107 KB staged; 2 blocks/WGP within 320 KB LDS

struct LevelParams {
  const float2* table;  // global table base (rows of 2 floats)
  int   res;            // grid resolution
  int   hash;           // 1 => hashed indexing (res*res >= T)
  int   lds_off;        // byte offset into LDS staging buffer (staged levels)
  int   entries;        // number of (f0,f1) rows in table
  float grid;           // 2048 / res in float32 (matches reference)
};
struct Params { LevelParams L[N_LEVELS]; };

__global__ __launch_bounds__(BLOCK)
void hashgrid_fwd(const float2* __restrict__ xs, float* __restrict__ out, Params P)
{
  __shared__ __align__(16) unsigned char smem[SMEM_BYTES];
  const int tid = threadIdx.x;

  // ---- Stage coarse-level tables into LDS with gfx1250 async global->LDS DMA.
#pragma unroll
  for (int L = 0; L < N_STAGED; ++L) {
    const int off = P.L[L].lds_off;
    const int n  = P.L[L].entries;                    // 8 bytes per row
    const float2* __restrict__ src = P.L[L].table;
    float2* dst = (float2*)(smem + off);
    for (int i = tid; i < n; i += BLOCK) {
      // Low 32 bits of a flat LDS pointer are the LDS byte address (ISA §10.2).
      unsigned lds_addr = (unsigned)(unsigned long long)(const void*)(dst + i);
      unsigned long long gaddr = (unsigned long long)(const void*)(src + i);
      asm volatile("global_load_async_to_lds_b64 %0, %1, off"
                   :: "v"(lds_addr), "v"(gaddr) : "memory");
    }
  }
  asm volatile("s_wait_asynccnt 0x0" ::: "memory");
  __syncthreads();

  const int base = blockIdx.x * (BLOCK * PPT);
#pragma unroll 1
  for (int k = 0; k < PPT; ++k) {
    const int p = base + k * BLOCK + tid;
    const float2 xy = xs[p];                          // global_load_b64, coalesced
    float acc[2 * N_LEVELS];

    // ---- Levels 0..5: LDS-resident tables, direct (non-hash) indexing.
#pragma unroll
    for (int L = 0; L < N_STAGED; ++L) {
      const float g  = P.L[L].grid;
      const int  res = P.L[L].res;
      const float gx = xy.x / g;
      const float gy = xy.y / g;
      const float fx = floorf(gx);
      const float fy = floorf(gy);
      const float w0 = gx - fx;
      const float w1 = gy - fy;
      const int bx = (int)fx, by = (int)fy;
      const int i00 = bx * res + by;
      const int i10 = i00 + res;
      const float2* t = (const float2*)(smem + P.L[L].lds_off);   // AS3 only
      const float2 e00 = t[i00];                      // ds_load_b64
      const float2 e01 = t[i00 + 1];
      const float2 e10 = t[i10];
      const float2 e11 = t[i10 + 1];
      const float om1 = 1.0f - w1, om0 = 1.0f - w0;
      const float c0x = e00.x * om1 + e01.x * w1;
      const float c0y = e00.y * om1 + e01.y * w1;
      const float c1x = e10.x * om1 + e11.x * w1;
      const float c1y = e10.y * om1 + e11.y * w1;
      acc[2 * L + 0] = c0x * om0 + c1x * w0;
      acc[2 * L + 1] = c0y * om0 + c1y * w0;
    }

    // ---- Levels 6..15: L2-resident tables; levels 12..15 use hashing.
#pragma unroll
    for (int L = N_STAGED; L < N_LEVELS; ++L) {
      const float g  = P.L[L].grid;
      const int  res = P.L[L].res;
      const float gx = xy.x / g;
      const float gy = xy.y / g;
      const float fx = floorf(gx);
      const float fy = floorf(gy);
      const float w0 = gx - fx;
      const float w1 = gy - fy;
      int i00, i01, i10, i11;
      if (P.L[L].hash) {                              // uniform per level
        const unsigned ux  = (unsigned)(int)fx;
        const unsigned uy  = (unsigned)(int)fy;
        const unsigned hy0 = uy * 2654435761u;
        const unsigned hy1 = (uy + 1u) * 2654435761u;
        i00 = (int)((ux        ^ hy0) & TMASK);
        i01 = (int)((ux        ^ hy1) & TMASK);
        i10 = (int)(((ux + 1u) ^ hy0) & TMASK);
        i11 = (int)(((ux + 1u) ^ hy1) & TMASK);
      } else {
        const int bx = (int)fx, by = (int)fy;
        i00 = bx * res + by; i01 = i00 + 1; i10 = i00 + res; i11 = i10 + 1;
      }
      const float2* __restrict__ t = P.L[L].table;    // AS1 only
      const float2 e00 = t[i00];                      // global_load_b64 (L2)
      const float2 e01 = t[i01];
      const float2 e10 = t[i10];
      const float2 e11 = t[i11];
      const float om1 = 1.0f - w1, om0 = 1.0f - w0;
      const float c0x = e00.x * om1 + e01.x * w1;
      const float c0y = e00.y * om1 + e01.y * w1;
      const float c1x = e10.x * om1 + e11.x * w1;
      const float c1y = e10.y * om1 + e11.y * w1;
      acc[2 * L + 0] = c0x * om0 + c1x * w0;
      acc[2 * L + 1] = c0y * om0 + c1y * w0;
    }

    // 128 contiguous bytes per point -> 8x global_store_b128
    float4* o = (float4*)(out + (size_t)p * (2 * N_LEVELS));
#pragma unroll
    for (int j = 0; j < 8; ++j)
      o[j] = make_float4(acc[4 * j + 0], acc[4 * j + 1], acc[4 * j + 2], acc[4 * j + 3]);
  }
}

extern "C" void kernel_launch(void* const* d_in, const int* in_sizes, int n_in,
                              void* d_out, int out_size, void* d_ws, size_t ws_size,
                              hipStream_t stream) {
  (void)d_ws; (void)ws_size; (void)n_in; (void)out_size;
  Params P;
  // Replicate Python: B = exp((log(2048)-log(16))/15); res_i = floor(16 * B**i)
  const double B = exp((log(2048.0) - log(16.0)) / 15.0);
  int off = 0;
  for (int i = 0; i < N_LEVELS; ++i) {
    const int res     = (int)floor(16.0 * pow(B, (double)i));
    const int entries = in_sizes[1 + i] / 2;          // min((res+1)^2, T) rows
    P.L[i].table   = (const float2*)d_in[1 + i];
    P.L[i].res     = res;
    P.L[i].hash    = ((long long)res * res < (1ll << LOG2T)) ? 0 : 1;
    P.L[i].grid    = 2048.0f / (float)res;            // float32 division, as reference
    P.L[i].entries = entries;
    if (i < N_STAGED) { P.L[i].lds_off = off; off += (entries * 8 + 15) & ~15; }
    else              { P.L[i].lds_off = -1; }
  }
  const int n_points = in_sizes[0] / 2;               // 1048576
  const int grid     = n_points / (BLOCK * PPT);      // 1024 blocks of 256 (8 waves)
  hashgrid_fwd<<<grid, BLOCK, 0, stream>>>((const float2*)d_in[0], (float*)d_out, P);
}